// VAELoss_5248450035913
// MI455X (gfx1250) — compile-verified
//
#include <hip/hip_runtime.h>
#include <math.h>

#define NROWS   8192
#define VSIZE   32000
#define NCHUNK  (VSIZE / 4)      // 8000 float4 chunks per row
#define TPB     256
#define TILES   ((NCHUNK + TPB - 1) / TPB)   // 32 tiles (last partial)
#define LATENT  64

// ---------------------------------------------------------------------------
// Async global -> LDS copy (CDNA5 path, ASYNCcnt-tracked)
// ---------------------------------------------------------------------------
__device__ __forceinline__ void async_copy_b128(unsigned lds_off,
                                                unsigned long long gaddr) {
    // GV mode: VDST = per-lane LDS byte offset, VADDR = per-lane 64-bit address
    asm volatile("global_load_async_to_lds_b128 %0, %1, off"
                 :: "v"(lds_off), "v"(gaddr) : "memory");
}
__device__ __forceinline__ void wait_async_le1() {
    asm volatile("s_wait_asynccnt 0x1" ::: "memory");
}
__device__ __forceinline__ void wait_async_le0() {
    asm volatile("s_wait_asynccnt 0x0" ::: "memory");
}

// Combine two (m, s) online-logsumexp states; safe for m == -inf, s == 0.
__device__ __forceinline__ void lse_combine(float& m, float& s, float mo, float so) {
    float nm = fmaxf(m, mo);
    float em = (m  == nm) ? 1.0f : __expf(m  - nm);
    float eo = (mo == nm) ? 1.0f : __expf(mo - nm);
    s = s * em + so * eo;
    m = nm;
}

// ---------------------------------------------------------------------------
// Kernel 1: per-row streaming logsumexp + target-logit gather.
// One block per row; row data double-buffered through LDS with async copies.
// rowNll[row] = logsumexp(x[row,:]) - x[row, y_true[row]]
// ---------------------------------------------------------------------------
__global__ __launch_bounds__(TPB)
void vae_row_lse(const float* __restrict__ x, const int* __restrict__ ytrue,
                 float* __restrict__ rowNll) {
    const int row = blockIdx.x;
    const int tid = threadIdx.x;
    const float* rowp = x + (size_t)row * VSIZE;

    const int ty     = ytrue[row];
    const int tchunk = ty >> 2;
    const int tsub   = ty & 3;

    __shared__ float4 buf[2][TPB];     // 2 x 4KB double buffer
    __shared__ float  red_m[8], red_s[8];
    __shared__ float  s_xt;

    auto issue = [&](int t) {
        int gc = t * TPB + tid;
        gc = (gc < NCHUNK) ? gc : (NCHUNK - 1);          // clamp: always issue
        unsigned loff = (unsigned)(size_t)(void*)&buf[t & 1][tid];
        unsigned long long ga = (unsigned long long)(size_t)(rowp + (size_t)gc * 4);
        async_copy_b128(loff, ga);
    };

    issue(0);

    float m = -INFINITY, s = 0.0f, xt = 0.0f;
    bool  found = false;

    for (int t = 0; t < TILES; ++t) {
        if (t + 1 < TILES) { issue(t + 1); wait_async_le1(); }
        else               { wait_async_le0(); }

        const int gc = t * TPB + tid;
        float4 v = buf[t & 1][tid];                      // ds_load_b128
        if (gc < NCHUNK) {
            float e0 = v.x, e1 = v.y, e2 = v.z, e3 = v.w;
            if (gc == tchunk) {
                xt = (tsub == 0) ? e0 : (tsub == 1) ? e1 : (tsub == 2) ? e2 : e3;
                found = true;
            }
            float elem[4] = {e0, e1, e2, e3};
            #pragma unroll
            for (int k = 0; k < 4; ++k) {
                float xv = elem[k];
                // one exp per element: d = exp(-|xv - m|)
                float d  = __expf(fminf(xv, m) - fmaxf(xv, m));
                s = (xv > m) ? __fmaf_rn(s, d, 1.0f) : (s + d);
                m = fmaxf(m, xv);
            }
        }
    }

    // wave32 reduction of (m, s)
    #pragma unroll
    for (int off = 16; off > 0; off >>= 1) {
        float mo = __shfl_xor(m, off, 32);
        float so = __shfl_xor(s, off, 32);
        lse_combine(m, s, mo, so);
    }

    const int wave = tid >> 5, lane = tid & 31;
    if (lane == 0) { red_m[wave] = m; red_s[wave] = s; }
    if (found) s_xt = xt;                                 // exactly one writer
    __syncthreads();

    if (tid == 0) {
        float M = red_m[0], S = red_s[0];
        #pragma unroll
        for (int w = 1; w < 8; ++w) lse_combine(M, S, red_m[w], red_s[w]);
        rowNll[row] = (M + __logf(S)) - s_xt;             // = -logp[target]
    }
}

// ---------------------------------------------------------------------------
// Kernel 2: 64 blocks; each reduces KLD terms for 128 rows (8192 elements)
// and the matching rowNll slice. Deterministic LDS tree reduction.
// ---------------------------------------------------------------------------
__global__ __launch_bounds__(TPB)
void vae_kld_partial(const float* __restrict__ mu, const float* __restrict__ lv,
                     const float* __restrict__ rowNll, float* __restrict__ parts) {
    const int b = blockIdx.x, tid = threadIdx.x;
    const int base = b * 128 * LATENT;                   // 8192 elements / block

    float acc = 0.0f;
    for (int i = tid; i < 128 * LATENT; i += TPB) {
        float mq = mu[base + i], l = lv[base + i];
        acc += 0.5f * (__expf(l) - l + mq * mq - 1.0f);
    }
    float nll = 0.0f;
    if (tid < 128) nll = rowNll[b * 128 + tid];

    __shared__ float sa[TPB], sn[TPB];
    sa[tid] = acc; sn[tid] = nll;
    __syncthreads();
    for (int st = TPB / 2; st > 0; st >>= 1) {
        if (tid < st) { sa[tid] += sa[tid + st]; sn[tid] += sn[tid + st]; }
        __syncthreads();
    }
    if (tid == 0) { parts[b] = sa[0]; parts[64 + b] = sn[0]; }
}

// ---------------------------------------------------------------------------
// Kernel 3: final scalar, fixed-order combine (deterministic)
// ---------------------------------------------------------------------------
__global__ void vae_finalize(const float* __restrict__ parts,
                             const float* __restrict__ anneal,
                             float* __restrict__ out) {
    if (threadIdx.x == 0 && blockIdx.x == 0) {
        float k = 0.0f, nl = 0.0f;
        for (int i = 0; i < 64; ++i) { k += parts[i]; nl += parts[64 + i]; }
        out[0] = anneal[0] * (k / (float)NROWS) + (nl / (float)NROWS);
    }
}

// ---------------------------------------------------------------------------
extern "C" void kernel_launch(void* const* d_in, const int* in_sizes, int n_in,
                              void* d_out, int out_size, void* d_ws, size_t ws_size,
                              hipStream_t stream) {
    const float* dec    = (const float*)d_in[0];   // [1, 8192, 32000] f32
    const float* mu     = (const float*)d_in[1];   // [8192, 64] f32
    const float* logvar = (const float*)d_in[2];   // [8192, 64] f32
    const int*   ytrue  = (const int*)  d_in[3];   // [8192] i32
    const float* anneal = (const float*)d_in[4];   // scalar f32
    float*       out    = (float*)d_out;

    float* rowNll = (float*)d_ws;                  // 8192 floats
    float* parts  = rowNll + NROWS;                // 128 floats (64 kld, 64 nll)

    vae_row_lse   <<<NROWS, TPB, 0, stream>>>(dec, ytrue, rowNll);
    vae_kld_partial<<<64,   TPB, 0, stream>>>(mu, logvar, rowNll, parts);
    vae_finalize  <<<1,      32, 0, stream>>>(parts, anneal, out);
}